// AttentionLayer_65377992180134
// MI455X (gfx1250) — compile-verified
//
#include <hip/hip_runtime.h>
#include <hip/hip_bf16.h>
#include <math.h>

typedef __attribute__((ext_vector_type(2))) float        v2f;
typedef __attribute__((ext_vector_type(8))) float        v8f;
typedef __attribute__((ext_vector_type(4))) unsigned int v4u;
typedef __attribute__((ext_vector_type(8))) int          v8i;
typedef __attribute__((ext_vector_type(4))) int          v4i;

#define B_  64
#define S_  2048
#define F_  512
#define A_  128
#define KC  64                        // K-chunk (32 KB staging buffer each)
#define NCHUNK (F_ / KC)              // 8
#define FLOAT32_MIN_ (-3.4028234663852886e38f)

// ---------------------------------------------------------------------------
// TDM: DMA one contiguous KC x A_ fp32 chunk of W into LDS (row-major).
// D# per CDNA5 ISA ch.8: group0 = count/lds_addr/global_addr/type,
// group1 = data_size, tensor dims, tile dims, dim0 stride. 2-D tensor ->
// remaining groups zero.  Issued wave-uniform; tracked with TENSORcnt.
// ---------------------------------------------------------------------------
__device__ __forceinline__ void tdm_load_chunk(const float* gsrc, void* lds_dst)
{
    unsigned long long ga  = (unsigned long long)(size_t)gsrc;
    unsigned int       lds = (unsigned int)(size_t)lds_dst; // low 32b = LDS offset

    v4u g0;
    g0[0] = 1u;                                             // count=1 (valid), user mode
    g0[1] = lds;                                            // lds_addr [63:32]
    g0[2] = (unsigned int)(ga & 0xFFFFFFFFu);               // global_addr [95:64]
    g0[3] = (unsigned int)((ga >> 32) & 0x1FFFFFFu)         // global_addr [120:96]
          | (2u << 30);                                     // type=2 ("image")

    v8i g1;
    g1[0] = (int)(2u << 16);      // wg_mask=0, data_size=2 (4 bytes)
    g1[1] = (int)(A_ << 16);      // tensor_dim0[15:0] = 128 (bits 63:48)
    g1[2] = (int)(KC << 16);      // tensor_dim0 hi=0, tensor_dim1[15:0] = 64
    g1[3] = (int)(A_ << 16);      // tensor_dim1 hi=0, tile_dim0 = 128
    g1[4] = (int)KC;              // tile_dim1 = 64, tile_dim2 = 0
    g1[5] = (int)A_;              // tensor_dim0_stride = 128 (elements)
    g1[6] = 0;                    // stride hi / tensor_dim1_stride lo (unused, 2-D)
    g1[7] = 0;

    v4i g2 = {0, 0, 0, 0};        // no dim2/3, no iterate
    v4i g3 = {0, 0, 0, 0};
    v8i g4 = {0, 0, 0, 0, 0, 0, 0, 0};

    __builtin_amdgcn_tensor_load_to_lds(g0, g1, g2, g3, g4, 0);
}

// ---------------------------------------------------------------------------
// Kernel 1: ait[b,s] = mask ? u . tanh(X[b,s,:] @ W + b) : FLOAT32_MIN
// One wave = one 16-row M-tile; 8 N-tiles of 16x16 accumulated with
// V_WMMA_F32_16X16X4_F32.  W chunks DMA'd by the Tensor Data Mover into
// double-buffered staging LDS, then repacked in-LDS to the paired-K layout
// (fragment = one aligned ds_load_b64, no register shuffles).
// ---------------------------------------------------------------------------
__global__ __launch_bounds__(256) void k_gemm_tanh_dot(
    const float* __restrict__ X,            // [B*S, F]
    const unsigned char* __restrict__ mask, // [B*S]
    const float* __restrict__ W,            // [F, A]
    const float* __restrict__ bias,         // [A]
    const float* __restrict__ u,            // [A]
    float* __restrict__ ait)                // [B*S]
{
    // sP declared FIRST so it lands at LDS offset 0: every fragment address
    // then fits the 16-bit DS immediate-offset field (no per-load v_add).
    __shared__ float2 sP[(KC / 2) * A_];    // 32 KB paired-K compute buffer
    __shared__ float  sStage[2][KC][A_];    // 2 x 32 KB (TDM destination)

    const int tid  = threadIdx.x;
    const int lane = tid & 31;
    const int wave = tid >> 5;              // 0..7
    const int half = lane >> 4;             // 0 or 1 (K sub-group)
    const int l16  = lane & 15;             // M (for A) / N (for B)

    const int mtile = blockIdx.x * 8 + wave;   // 0..8191
    const int r0    = mtile * 16;

    // A-fragment base: row r0+l16, starting at K-offset 2*half
    const float* aptr = X + (size_t)(r0 + l16) * F_ + 2 * half;

    v8f acc[8] = {};   // 8 N-tiles of 16x16 f32 C/D

    if (wave == 0)
        tdm_load_chunk(W, &sStage[0][0][0]);    // prefetch chunk 0

    #pragma unroll 1
    for (int c = 0; c < NCHUNK; ++c) {
        if (wave == 0) {
            if (c + 1 < NCHUNK) {           // DMA next chunk into other buffer
                tdm_load_chunk(W + (size_t)(c + 1) * KC * A_,
                               &sStage[(c + 1) & 1][0][0]);
                __builtin_amdgcn_s_wait_tensorcnt(1);  // chunk c resident (in-order)
            } else {
                __builtin_amdgcn_s_wait_tensorcnt(0);
            }
        }
        __syncthreads();   // staging c ready; previous compute done (sP free)

        // --- repack staging (row-major) -> paired-K: sP[p][n] = {W[2p][n], W[2p+1][n]}
        // ds_load_2addr_b32 lands the pair in (VDST, VDST+1); stored as one b64.
        {
            const float* st = &sStage[c & 1][0][0];
            #pragma unroll
            for (int i = 0; i < (KC * A_ / 2) / 256; ++i) {   // 16 pairs/thread
                int idx = i * 256 + tid;                      // pair index
                int p   = idx >> 7;                           // 0..KC/2-1
                int n   = idx & (A_ - 1);
                float2 v;
                v.x = st[(2 * p) * A_ + n];
                v.y = st[(2 * p) * A_ + A_ + n];
                sP[idx] = v;
            }
        }
        __syncthreads();

        const float* abase = aptr + c * KC;

        #pragma unroll 4
        for (int ks = 0; ks < KC / 4; ++ks) {
            float2 av = *(const float2*)(abase + ks * 4);   // global_load_b64
            v2f afrag; afrag.x = av.x; afrag.y = av.y;

            const float2* brow = sP + (2 * ks + half) * A_;
            #pragma unroll
            for (int j = 0; j < 8; ++j) {
                float2 bv = brow[j * 16 + l16];             // ds_load_b64 (imm offset)
                v2f bfrag; bfrag.x = bv.x; bfrag.y = bv.y;
                acc[j] = __builtin_amdgcn_wmma_f32_16x16x4_f32(
                    false, afrag, false, bfrag, (short)0, acc[j], false, false);
            }
        }
        __syncthreads();
    }

    // --- epilogue: s[row] = sum_a tanh(acc + bias[a]) * u[a]
    float bj[8], uj[8];
    #pragma unroll
    for (int j = 0; j < 8; ++j) {
        bj[j] = bias[j * 16 + l16];
        uj[j] = u[j * 16 + l16];
    }

    #pragma unroll
    for (int g = 0; g < 8; ++g) {           // VGPR index g <-> row g + 8*half
        float s = 0.0f;
        #pragma unroll
        for (int j = 0; j < 8; ++j)
            s += tanhf(acc[j][g] + bj[j]) * uj[j];
        // reduce the 16 lanes of each half-wave (rows g and g+8)
        s += __shfl_xor(s, 1, 32);
        s += __shfl_xor(s, 2, 32);
        s += __shfl_xor(s, 4, 32);
        s += __shfl_xor(s, 8, 32);
        if (l16 == g) {
            int row = r0 + g + 8 * half;
            ait[row] = mask[row] ? s : FLOAT32_MIN_;
        }
    }
}

// ---------------------------------------------------------------------------
// Kernel 2: softmax over S per batch row.  64 blocks x 256 threads, 8 el/thr.
// ---------------------------------------------------------------------------
__global__ __launch_bounds__(256) void k_softmax(
    const float* __restrict__ ait, float* __restrict__ aw)
{
    __shared__ float red[8];
    const int b   = blockIdx.x;
    const int tid = threadIdx.x;
    const int lane = tid & 31, wid = tid >> 5;
    const float* row = ait + (size_t)b * S_;

    float v[8];
    float m = FLOAT32_MIN_;
    #pragma unroll
    for (int i = 0; i < 8; ++i) {
        v[i] = row[tid + i * 256];
        m = fmaxf(m, v[i]);
    }
    #pragma unroll
    for (int off = 16; off; off >>= 1) m = fmaxf(m, __shfl_xor(m, off, 32));
    if (lane == 0) red[wid] = m;
    __syncthreads();
    float mm = red[0];
    #pragma unroll
    for (int i = 1; i < 8; ++i) mm = fmaxf(mm, red[i]);
    __syncthreads();

    float s = 0.0f;
    #pragma unroll
    for (int i = 0; i < 8; ++i) { v[i] = __expf(v[i] - mm); s += v[i]; }
    #pragma unroll
    for (int off = 16; off; off >>= 1) s += __shfl_xor(s, off, 32);
    if (lane == 0) red[wid] = s;
    __syncthreads();
    float ss = 0.0f;
    #pragma unroll
    for (int i = 0; i < 8; ++i) ss += red[i];
    float inv = 1.0f / ss;

    #pragma unroll
    for (int i = 0; i < 8; ++i)
        aw[(size_t)b * S_ + tid + i * 256] = v[i] * inv;
}

// ---------------------------------------------------------------------------
// Kernel 3: out[b,f] = sum_s X[b,s,f] * aw[b,s].  Pure HBM-bandwidth pass.
// 128 blocks: block -> (b, 256-wide f chunk); aw row staged in LDS.
// ---------------------------------------------------------------------------
__global__ __launch_bounds__(256) void k_weighted_sum(
    const float* __restrict__ X, const float* __restrict__ aw,
    float* __restrict__ out)
{
    __shared__ float saw[S_];               // 8 KB
    const int b = blockIdx.x >> 1;
    const int f = ((blockIdx.x & 1) << 8) + threadIdx.x;

    #pragma unroll
    for (int i = 0; i < 8; ++i)
        saw[threadIdx.x + i * 256] = aw[(size_t)b * S_ + threadIdx.x + i * 256];
    __syncthreads();

    const float* xp = X + (size_t)b * S_ * F_ + f;
    float acc = 0.0f;
    #pragma unroll 8
    for (int s = 0; s < S_; ++s)
        acc = fmaf(xp[(size_t)s * F_], saw[s], acc);

    out[b * F_ + f] = acc;
}

// ---------------------------------------------------------------------------
extern "C" void kernel_launch(void* const* d_in, const int* in_sizes, int n_in,
                              void* d_out, int out_size, void* d_ws, size_t ws_size,
                              hipStream_t stream)
{
    const float*         X    = (const float*)d_in[0];         // [B,S,F]
    const unsigned char* mask = (const unsigned char*)d_in[1]; // [B,S] bool
    const float*         W    = (const float*)d_in[2];         // [F,A]
    const float*         bias = (const float*)d_in[3];         // [A]
    const float*         u    = (const float*)d_in[4];         // [A]

    float* out = (float*)d_out;          // [B,F]
    float* aw  = out + B_ * F_;          // [B,S,1] (second tuple element)
    float* ait = (float*)d_ws;           // [B,S] scratch (512 KB)

    k_gemm_tanh_dot<<<(B_ * S_) / (16 * 8), 256, 0, stream>>>(X, mask, W, bias, u, ait);
    k_softmax<<<B_, 256, 0, stream>>>(ait, aw);
    k_weighted_sum<<<B_ * (F_ / 256), 256, 0, stream>>>(X, aw, out);
}